// CBOWModel_63196148793603
// MI455X (gfx1250) — compile-verified
//
#include <hip/hip_runtime.h>

typedef float v2f __attribute__((ext_vector_type(2)));
typedef float v8f __attribute__((ext_vector_type(8)));

#define V_SZ  100000
#define D_DIM 128
#define B_SZ  16384
#define C_CTX 8
#define K_NEG 5
#define NSCORE (1 + K_NEG)   // 1 positive + 5 negatives

// Numerically stable log(sigmoid(x)) = min(x,0) - log1p(exp(-|x|))
__device__ __forceinline__ float log_sigmoid(float x) {
    return fminf(x, 0.0f) - log1pf(__expf(-fabsf(x)));
}

// Select element v (0..7) from a v8f accumulator without dynamic VGPR indexing.
__device__ __forceinline__ float diag_sel(v8f c, int v) {
    float r = c[0];
    r = (v == 1) ? c[1] : r;
    r = (v == 2) ? c[2] : r;
    r = (v == 3) ? c[3] : r;
    r = (v == 4) ? c[4] : r;
    r = (v == 5) ? c[5] : r;
    r = (v == 6) ? c[6] : r;
    r = (v == 7) ? c[7] : r;
    return r;
}

__global__ void cbow_zero_out(float* out) {
    if (threadIdx.x == 0) out[0] = 0.0f;
}

// One wave (32 lanes) handles a tile of 16 batch rows.
// Scores for the tile = diag( U(16x128) x W(16x128)^T ) via 32 chained
// v_wmma_f32_16x16x4_f32 ops per score matrix (6 matrices: 1 pos + 5 neg).
__global__ __launch_bounds__(256)
void cbow_loss_kernel(const int* __restrict__ pos_u,    // [B, C]
                      const int* __restrict__ pos_w,    // [B]
                      const int* __restrict__ neg_w,    // [B, K]
                      const float* __restrict__ u_emb,  // [V, D]
                      const float* __restrict__ w_emb,  // [V, D]
                      float* __restrict__ out)          // [1]
{
    const int lane = threadIdx.x & 31;
    const int wave = threadIdx.x >> 5;
    const int tile = blockIdx.x * 8 + wave;   // 16-batch tile index
    const int row  = lane & 15;               // matrix row (A) / col (B) owned by this lane
    const int off  = (lane >> 4) << 1;        // k sub-offset within each 4-chunk: 0 or 2
    const int b    = tile * 16 + row;         // batch element this lane gathers for

    // ---- Build A fragments: pos_u_emb row = sum of C=8 context embeddings ----
    // a[j] holds A[row][4*j + off .. 4*j + off + 1]  (ISA 16x4 f32 A layout)
    v2f a[32];
#pragma unroll
    for (int j = 0; j < 32; ++j) a[j] = (v2f)0.0f;

    for (int c = 0; c < C_CTX; ++c) {
        const float* urow = u_emb + (size_t)pos_u[b * C_CTX + c] * D_DIM;
#pragma unroll
        for (int j = 0; j < 32; ++j) {
            v2f v = *(const v2f*)(urow + j * 4 + off);
            a[j] += v;
        }
    }

    // ---- w_emb row pointers: B = W^T, lane streams row (lane&15) ----
    const float* wrow[NSCORE];
    wrow[0] = w_emb + (size_t)pos_w[b] * D_DIM;
#pragma unroll
    for (int k = 0; k < K_NEG; ++k)
        wrow[k + 1] = w_emb + (size_t)neg_w[b * K_NEG + k] * D_DIM;

    v8f acc[NSCORE];
#pragma unroll
    for (int s = 0; s < NSCORE; ++s) acc[s] = (v8f)0.0f;

    // ---- 32 k-chunks x 6 score matrices of v_wmma_f32_16x16x4_f32 ----
#pragma unroll
    for (int j = 0; j < 32; ++j) {
#pragma unroll
        for (int s = 0; s < NSCORE; ++s) {
            v2f bfrag = *(const v2f*)(wrow[s] + j * 4 + off);
            acc[s] = __builtin_amdgcn_wmma_f32_16x16x4_f32(
                /*neg_a=*/false, a[j],
                /*neg_b=*/false, bfrag,
                /*c_mod=*/(short)0, acc[s],
                /*reuse_a=*/false, /*reuse_b=*/false);
        }
    }

    // ---- Diagonal extraction + log-sigmoid ----
    // (m,m) lives in: lanes 0..7 at vgpr m, lanes 24..31 at vgpr m-8.
    float contrib = 0.0f;
    const bool is_diag = (lane < 8) || (lane >= 24);
    const int  v       = (lane < 8) ? lane : (lane - 24);
    if (is_diag) {
        contrib = log_sigmoid(diag_sel(acc[0], v));          // positive term
#pragma unroll
        for (int s = 1; s < NSCORE; ++s)
            contrib += log_sigmoid(-diag_sel(acc[s], v));    // negative terms
    }

    // ---- wave32 reduction, one atomic per wave ----
#pragma unroll
    for (int m = 16; m >= 1; m >>= 1)
        contrib += __shfl_xor(contrib, m, 32);
    if (lane == 0)
        atomicAdd(out, -contrib);
}

extern "C" void kernel_launch(void* const* d_in, const int* in_sizes, int n_in,
                              void* d_out, int out_size, void* d_ws, size_t ws_size,
                              hipStream_t stream) {
    (void)in_sizes; (void)n_in; (void)out_size; (void)d_ws; (void)ws_size;
    const int*   pos_u = (const int*)d_in[0];
    const int*   pos_w = (const int*)d_in[1];
    const int*   neg_w = (const int*)d_in[2];
    const float* u_emb = (const float*)d_in[3];
    const float* w_emb = (const float*)d_in[4];
    float* out = (float*)d_out;

    cbow_zero_out<<<1, 32, 0, stream>>>(out);
    // 8 waves/block, each wave = 16 batches -> 128 batches/block
    cbow_loss_kernel<<<B_SZ / 128, 256, 0, stream>>>(
        pos_u, pos_w, neg_w, u_emb, w_emb, out);
}